// FullPNAAggregator_25984552141031
// MI455X (gfx1250) — compile-verified
//
#include <hip/hip_runtime.h>
#include <math.h>

#define N_NODES 50000
#define D_FEAT  128
#define IN_DIM  (13 * D_FEAT)   // 1664
#define TILE_STRIDE 132         // 128 + 4 pad: A-frag LDS reads hit distinct banks

typedef float v2f __attribute__((ext_vector_type(2)));
typedef float v8f __attribute__((ext_vector_type(8)));

// ---- order-preserving uint encoding for float max/min via integer atomics ----
__device__ __forceinline__ unsigned enc_f32(float f) {
    unsigned u = __float_as_uint(f);
    return (u & 0x80000000u) ? ~u : (u | 0x80000000u);
}
__device__ __forceinline__ float dec_f32(unsigned e) {
    unsigned u = (e & 0x80000000u) ? (e ^ 0x80000000u) : ~e;
    return __uint_as_float(u);
}

// ---------------- kernel 1: init workspace ----------------
__global__ void pna_init_kernel(float* __restrict__ sum, float* __restrict__ sumsq,
                                unsigned* __restrict__ maxe, unsigned* __restrict__ mine,
                                unsigned* __restrict__ deg, float* __restrict__ logsum,
                                size_t nd, int n) {
    size_t i = (size_t)blockIdx.x * blockDim.x + threadIdx.x;
    if (i < nd) {
        sum[i]   = 0.0f;
        sumsq[i] = 0.0f;
        maxe[i]  = 0u;           // encoded -inf (smallest encoding)
        mine[i]  = 0xFFFFFFFFu;  // encoded +inf (largest encoding)
    }
    if (i < (size_t)n) deg[i] = 0u;
    if (i == 0) *logsum = 0.0f;
}

// ---------------- kernel 2: edge scatter (1 wave per edge) ----------------
__global__ void pna_scatter_kernel(const float* __restrict__ msgs, const int* __restrict__ dst,
                                   float* __restrict__ sum, float* __restrict__ sumsq,
                                   unsigned* __restrict__ maxe, unsigned* __restrict__ mine,
                                   unsigned* __restrict__ deg, int E) {
    long t = (long)blockIdx.x * blockDim.x + threadIdx.x;
    int e = (int)(t >> 5);
    if (e >= E) return;
    int lane = (int)(t & 31);
    int d = dst[e];
    const float4 m4 = *(const float4*)(msgs + (size_t)e * D_FEAT + lane * 4);
    size_t base = (size_t)d * D_FEAT + lane * 4;
    float mv[4] = {m4.x, m4.y, m4.z, m4.w};
#pragma unroll
    for (int i = 0; i < 4; ++i) {
        atomicAdd(sum   + base + i, mv[i]);
        atomicAdd(sumsq + base + i, mv[i] * mv[i]);
        atomicMax(maxe  + base + i, enc_f32(mv[i]));
        atomicMin(mine  + base + i, enc_f32(mv[i]));
    }
    if (lane == 0) atomicAdd(deg + d, 1u);
}

// ---------------- kernel 3: sum of log(deg_clamped + 1) ----------------
__global__ void pna_degsum_kernel(const unsigned* __restrict__ deg, float* __restrict__ logsum, int n) {
    __shared__ float red[256];
    int i = blockIdx.x * 256 + threadIdx.x;
    float v = 0.0f;
    if (i < n) {
        float dc = fmaxf((float)deg[i], 1.0f);
        v = logf(dc + 1.0f);
    }
    red[threadIdx.x] = v;
    __syncthreads();
    for (int s = 128; s > 0; s >>= 1) {
        if (threadIdx.x < s) red[threadIdx.x] += red[threadIdx.x + s];
        __syncthreads();
    }
    if (threadIdx.x == 0) atomicAdd(logsum, red[0]);
}

// ---------------- kernel 4: transpose W -> Wt [128][1664] ----------------
__global__ void pna_wt_kernel(const float* __restrict__ W, float* __restrict__ Wt) {
    int i = blockIdx.x * 256 + threadIdx.x;
    if (i >= IN_DIM * D_FEAT) return;
    int k = i / D_FEAT, c = i % D_FEAT;
    Wt[(size_t)c * IN_DIM + k] = W[i];
}

// one feature group (128 K values): 32 steps of K=4, each B-frag feeds 2 WMMAs
__device__ __forceinline__ void pna_wmma_group(const float* __restrict__ at,
                                               const float* __restrict__ bt,
                                               v8f& c0, v8f& c1) {
#pragma unroll 4
    for (int k2 = 0; k2 < 128; k2 += 4) {
        float2 bv  = *(const float2*)(bt + k2);                      // global_load_b64 (L2-resident Wt)
        float2 a0v = *(const float2*)(at + k2);                      // ds_load_b64, rows 0-15
        float2 a1v = *(const float2*)(at + 16 * TILE_STRIDE + k2);   // ds_load_b64, rows 16-31
        v2f b;  b.x  = bv.x;  b.y  = bv.y;
        v2f a0; a0.x = a0v.x; a0.y = a0v.y;
        v2f a1; a1.x = a1v.x; a1.y = a1v.y;
        c0 = __builtin_amdgcn_wmma_f32_16x16x4_f32(false, a0, false, b, (short)0, c0, false, false);
        c1 = __builtin_amdgcn_wmma_f32_16x16x4_f32(false, a1, false, b, (short)0, c1, false, false);
    }
}

// ---------------- kernel 5: fused feature-build + WMMA GEMM ----------------
// block: 256 threads = 8 waves. Output tile: 32 nodes x 128 cols, 2 M-tiles/wave.
// All control flow is block-uniform => EXEC all-ones for every WMMA (tail handled
// by clamping staging reads and guarding only the final stores).
__global__ __launch_bounds__(256) void pna_gemm_kernel(
        const float* __restrict__ hprev, const float* __restrict__ Wt,
        const float* __restrict__ bias,
        const float* __restrict__ sum, const float* __restrict__ sumsq,
        const unsigned* __restrict__ maxe, const unsigned* __restrict__ mine,
        const unsigned* __restrict__ deg, const float* __restrict__ logsum,
        float* __restrict__ out) {
    __shared__ float sA[32 * TILE_STRIDE];   // one scaled 32x128 A-tile (16.9 KB)
    __shared__ float sBase[32], sInvB[32], sInvDeg[32], sHas[32];

    int tid = threadIdx.x;
    int nb  = blockIdx.x * 32;

    float delta = *logsum * (1.0f / (float)N_NODES);

    if (tid < 32) {
        int n = nb + tid; n = (n < N_NODES) ? n : (N_NODES - 1);
        float dcount = (float)deg[n];
        float dc = fmaxf(dcount, 1.0f);
        sInvDeg[tid] = 1.0f / dc;
        sHas[tid]    = (dcount > 0.0f) ? 1.0f : 0.0f;
        float base = fmaxf(logf(dc + 1.0f) / delta, 1e-6f);
        sBase[tid] = base;
        sInvB[tid] = 1.0f / base;
    }
    __syncthreads();

    // staging decomposition: thread owns column f, rows r0+2i (i=0..15)
    int f  = tid & 127;
    int r0 = tid >> 7;   // 0 or 1

    int wave  = tid >> 5;
    int lane  = tid & 31;
    int row   = lane & 15;   // A-matrix M row for this lane (ISA 16x4 f32 layout)
    int khalf = lane >> 4;   // K-pair select: {2h, 2h+1}
    int col   = 16 * wave + row;

    const float* at    = sA + row * TILE_STRIDE + 2 * khalf;
    const float* wtcol = Wt + (size_t)col * IN_DIM + 2 * khalf;

    v8f c0 = {}, c1 = {};

    // groups 0..11: stats {mean, std, max, min} x scales {1, base, 1/base}
    int gg = 0;
#pragma unroll 1
    for (int t = 0; t < 4; ++t) {
        float val[16];
#pragma unroll
        for (int i = 0; i < 16; ++i) {
            int r = r0 + 2 * i;
            int n = nb + r; n = (n < N_NODES) ? n : (N_NODES - 1);
            size_t g = (size_t)n * D_FEAT + f;
            float v;
            if (t == 0) {
                v = sum[g] * sInvDeg[r];
            } else if (t == 1) {
                float invd = sInvDeg[r];
                float mean = sum[g] * invd;
                float var  = fmaxf(sumsq[g] * invd - mean * mean, 0.0f);
                v = sqrtf(var + 1e-6f);
            } else if (t == 2) {
                v = dec_f32(maxe[g]) * sHas[r];
            } else {
                v = dec_f32(mine[g]) * sHas[r];
            }
            val[i] = v;
        }
#pragma unroll 1
        for (int m = 0; m < 3; ++m, ++gg) {
            __syncthreads();   // previous group's WMMA reads of sA are done
#pragma unroll
            for (int i = 0; i < 16; ++i) {
                int r = r0 + 2 * i;
                float s = (m == 0) ? 1.0f : ((m == 1) ? sBase[r] : sInvB[r]);
                sA[r * TILE_STRIDE + f] = val[i] * s;
            }
            __syncthreads();
            pna_wmma_group(at, wtcol + gg * 128, c0, c1);
        }
    }

    // group 12: h_prev (scale 1)
    __syncthreads();
#pragma unroll
    for (int i = 0; i < 16; ++i) {
        int r = r0 + 2 * i;
        int n = nb + r; n = (n < N_NODES) ? n : (N_NODES - 1);
        sA[r * TILE_STRIDE + f] = hprev[(size_t)n * D_FEAT + f];
    }
    __syncthreads();
    pna_wmma_group(at, wtcol + 12 * 128, c0, c1);

    // epilogue: C/D layout -> VGPR r holds M=r (lanes 0-15), M=r+8 (lanes 16-31)
    float bb = bias[col];
#pragma unroll
    for (int r = 0; r < 8; ++r) {
        int mrow = r + 8 * khalf;
        int n0 = nb + mrow;
        int n1 = nb + 16 + mrow;
        out[(size_t)n0 * D_FEAT + col] = c0[r] + bb;   // n0 always < N (nb+15 <= 49999)
        if (n1 < N_NODES) out[(size_t)n1 * D_FEAT + col] = c1[r] + bb;
    }
}

extern "C" void kernel_launch(void* const* d_in, const int* in_sizes, int n_in,
                              void* d_out, int out_size, void* d_ws, size_t ws_size,
                              hipStream_t stream) {
    const float* messages = (const float*)d_in[0];
    const int*   dst      = (const int*)d_in[1];   // per harness: integer -> const int*
    const float* hprev    = (const float*)d_in[2];
    const float* W        = (const float*)d_in[3];
    const float* b        = (const float*)d_in[4];
    int E = in_sizes[1];

    // workspace carve-out (~103.5 MB; all accumulators fit in MI455X's 192MB L2)
    char* ws = (char*)d_ws;
    size_t nd = (size_t)N_NODES * D_FEAT;
    float*    sum    = (float*)ws;    ws += nd * 4;
    float*    sumsq  = (float*)ws;    ws += nd * 4;
    unsigned* maxe   = (unsigned*)ws; ws += nd * 4;
    unsigned* mine   = (unsigned*)ws; ws += nd * 4;
    unsigned* deg    = (unsigned*)ws; ws += (size_t)N_NODES * 4;
    float*    logsum = (float*)ws;    ws += 16;
    float*    Wt     = (float*)ws;

    pna_init_kernel<<<(unsigned)((nd + 255) / 256), 256, 0, stream>>>(
        sum, sumsq, maxe, mine, deg, logsum, nd, N_NODES);

    long scatter_threads = (long)E * 32;
    pna_scatter_kernel<<<(unsigned)((scatter_threads + 255) / 256), 256, 0, stream>>>(
        messages, dst, sum, sumsq, maxe, mine, deg, E);

    pna_degsum_kernel<<<(N_NODES + 255) / 256, 256, 0, stream>>>(deg, logsum, N_NODES);

    pna_wt_kernel<<<(IN_DIM * D_FEAT + 255) / 256, 256, 0, stream>>>(W, Wt);

    int gemm_blocks = (N_NODES + 31) / 32;   // 1563 (tail block clamps reads, guards stores)
    pna_gemm_kernel<<<gemm_blocks, 256, 0, stream>>>(
        hprev, Wt, b, sum, sumsq, maxe, mine, deg, logsum, (float*)d_out);
}